// GATLayer_24068996727344
// MI455X (gfx1250) — compile-verified
//
#include <hip/hip_runtime.h>
#include <math.h>

typedef __attribute__((ext_vector_type(2))) float v2f;
typedef __attribute__((ext_vector_type(8))) float v8f;

#define HEADS 4
#define CPH   32
#define FDIM  128

// ---------- float atomic-max via ordered-uint encoding ----------
__device__ __forceinline__ unsigned int encF(float f) {
    unsigned int b = __float_as_uint(f);
    return (b & 0x80000000u) ? ~b : (b | 0x80000000u);
}
__device__ __forceinline__ float decF(unsigned int u) {
    return (u & 0x80000000u) ? __uint_as_float(u & 0x7FFFFFFFu)
                             : __uint_as_float(~u);
}

// ---------- K1: per-dst count + edge_attr sum (for self-loop fill 'mean') ----
__global__ void k_edge_cnt(const long long* __restrict__ dst,
                           const float* __restrict__ ea,
                           float* __restrict__ cnt, float* __restrict__ lsum,
                           int E) {
    int k = blockIdx.x * blockDim.x + threadIdx.x;
    if (k >= E) return;
    int d = (int)dst[k];
    atomicAdd(&cnt[d], 1.0f);
    atomicAdd(&lsum[d * 2 + 0], ea[k * 2 + 0]);
    atomicAdd(&lsum[d * 2 + 1], ea[k * 2 + 1]);
}

// ---------- K2: loop_attr = lsum / max(cnt,1), in place -------------------
__global__ void k_loop_attr(float* __restrict__ lsum,
                            const float* __restrict__ cnt, int N) {
    int n = blockIdx.x * blockDim.x + threadIdx.x;
    if (n >= N) return;
    float c = fmaxf(cnt[n], 1.0f);
    lsum[n * 2 + 0] /= c;
    lsum[n * 2 + 1] /= c;
}

// ---------- K3: p[j][h] = sum_c W_edge[j, h*32+c] * att_edge[h*32+c] ------
__global__ void k_pcoef(const float* __restrict__ We,
                        const float* __restrict__ att_e,
                        float* __restrict__ p) {
    int t = threadIdx.x;          // 0..7 -> (j,h)
    if (t >= 2 * HEADS) return;
    int j = t >> 2, h = t & 3;
    float s = 0.f;
    for (int c = 0; c < CPH; ++c)
        s += We[j * FDIM + h * CPH + c] * att_e[h * CPH + c];
    p[j * HEADS + h] = s;
}

// ---------- K4: h = x @ W via V_WMMA_F32_16X16X4_F32 ----------------------
// block = 256 threads (8 waves). Block handles 16 rows; wave w handles cols
// [16w, 16w+16). x tile (16x128) staged in LDS.
__global__ void __launch_bounds__(256)
k_gemm_wmma(const float* __restrict__ X, const float* __restrict__ W,
            float* __restrict__ H, int N) {
    __shared__ float sx[16 * FDIM];
    const int tid  = threadIdx.x;
    const int wave = tid >> 5;
    const int lane = tid & 31;
    const int half = lane >> 4;      // 0: K pair {0,1}, 1: K pair {2,3}
    const int lo   = lane & 15;
    const int row0 = blockIdx.x * 16;
    const int col0 = wave * 16;

    // stage 16x128 tile of X into LDS (clamp rows for generality)
    for (int i = tid; i < 16 * FDIM; i += 256) {
        int r = i >> 7, c = i & (FDIM - 1);
        int rr = row0 + r; if (rr >= N) rr = N - 1;
        sx[i] = X[(size_t)rr * FDIM + c];
    }
    __syncthreads();

    v8f acc = {};
    #pragma unroll
    for (int k = 0; k < FDIM; k += 4) {
        v2f a, b;
        // A 16x4: lanes 0-15 hold K=k,k+1 ; lanes 16-31 hold K=k+2,k+3
        a.x = sx[lo * FDIM + k + 2 * half + 0];
        a.y = sx[lo * FDIM + k + 2 * half + 1];
        // B 4x16: v0/v1 are rows K=k+2*half, K=k+2*half+1 striped over lanes
        const float* wp = W + (size_t)(k + 2 * half) * FDIM + col0 + lo;
        b.x = wp[0];
        b.y = wp[FDIM];
        acc = __builtin_amdgcn_wmma_f32_16x16x4_f32(
            false, a, false, b, (short)0, acc, false, false);
    }

    // C/D layout: VGPR r -> M = r (lanes 0-15) / M = 8+r (lanes 16-31), N = lo
    #pragma unroll
    for (int r = 0; r < 8; ++r) {
        int row = row0 + r + 8 * half;
        if (row < N) H[(size_t)row * FDIM + col0 + lo] = acc[r];
    }
}

// ---------- K5: a_src/a_dst dots, one thread per (node, head) -------------
__global__ void k_attdots(const float* __restrict__ H,
                          const float* __restrict__ att_s,
                          const float* __restrict__ att_d,
                          float* __restrict__ a_src,
                          float* __restrict__ a_dst, int N) {
    int t = blockIdx.x * blockDim.x + threadIdx.x;
    if (t >= N * HEADS) return;
    int n = t >> 2, h = t & 3;
    const float* hp = H + (size_t)n * FDIM + h * CPH;
    float ss = 0.f, sd = 0.f;
    #pragma unroll
    for (int c = 0; c < CPH; ++c) {
        float v = hp[c];
        ss += v * att_s[h * CPH + c];
        sd += v * att_d[h * CPH + c];
    }
    a_src[t] = ss;
    a_dst[t] = sd;
}

// ---------- K6: alpha = leakyrelu(a_src[s]+a_dst[d]+a_e), atomic max ------
__global__ void k_alpha_max(const long long* __restrict__ src,
                            const long long* __restrict__ dst,
                            const float* __restrict__ ea,
                            const float* __restrict__ loopa,
                            const float* __restrict__ a_src,
                            const float* __restrict__ a_dst,
                            const float* __restrict__ p,
                            float* __restrict__ alphaE,
                            unsigned int* __restrict__ mEnc,
                            int E, int K) {
    int k = blockIdx.x * blockDim.x + threadIdx.x;
    if (k >= K) return;
    int s_, d_;
    float e0, e1;
    if (k < E) {
        s_ = (int)src[k]; d_ = (int)dst[k];
        e0 = ea[k * 2 + 0]; e1 = ea[k * 2 + 1];
    } else {
        s_ = d_ = k - E;
        e0 = loopa[s_ * 2 + 0]; e1 = loopa[s_ * 2 + 1];
    }
    #pragma unroll
    for (int h = 0; h < HEADS; ++h) {
        float al = a_src[s_ * HEADS + h] + a_dst[d_ * HEADS + h]
                 + e0 * p[h] + e1 * p[HEADS + h];
        al = (al > 0.f) ? al : 0.2f * al;       // leaky_relu 0.2
        alphaE[(size_t)k * HEADS + h] = al;
        atomicMax(&mEnc[d_ * HEADS + h], encF(al));
    }
}

// ---------- K7: exp(alpha - max), atomic sum ------------------------------
__global__ void k_exp_sum(const long long* __restrict__ dst,
                          float* __restrict__ alphaE,
                          const unsigned int* __restrict__ mEnc,
                          float* __restrict__ ssum, int E, int K) {
    int k = blockIdx.x * blockDim.x + threadIdx.x;
    if (k >= K) return;
    int d_ = (k < E) ? (int)dst[k] : (k - E);
    #pragma unroll
    for (int h = 0; h < HEADS; ++h) {
        float m  = decF(mEnc[d_ * HEADS + h]);
        float ev = __expf(alphaE[(size_t)k * HEADS + h] - m);
        alphaE[(size_t)k * HEADS + h] = ev;
        atomicAdd(&ssum[d_ * HEADS + h], ev);
    }
}

// ---------- K8: weighted scatter: out[d] += w * h[s], 128 lanes / edge ----
__global__ void __launch_bounds__(128)
k_scatter(const long long* __restrict__ src,
          const long long* __restrict__ dst,
          const float* __restrict__ alphaE,
          const float* __restrict__ ssum,
          const float* __restrict__ H,
          float* __restrict__ out, int E) {
    int k = blockIdx.x;
    int c = threadIdx.x;              // 0..127
    int h = c >> 5;                   // head
    int s_, d_;
    if (k < E) { s_ = (int)src[k]; d_ = (int)dst[k]; }
    else       { s_ = d_ = k - E; }
    float w = alphaE[(size_t)k * HEADS + h]
            / (ssum[d_ * HEADS + h] + 1e-16f);
    atomicAdd(&out[(size_t)d_ * FDIM + c], w * H[(size_t)s_ * FDIM + c]);
}

// ---------- K9: BN stats (per-channel sum / sumsq) ------------------------
__global__ void __launch_bounds__(128)
k_bn_stats(const float* __restrict__ out, float* __restrict__ bsum,
           float* __restrict__ bsq, int N) {
    int c = threadIdx.x;
    float s = 0.f, q = 0.f;
    for (int n = blockIdx.x; n < N; n += gridDim.x) {
        float v = out[(size_t)n * FDIM + c];
        s += v; q += v * v;
    }
    atomicAdd(&bsum[c], s);
    atomicAdd(&bsq[c], q);
}

// ---------- K10: BN apply + leaky_relu(0.01), in place --------------------
__global__ void k_bn_apply(float* __restrict__ out,
                           const float* __restrict__ bsum,
                           const float* __restrict__ bsq,
                           const float* __restrict__ gamma,
                           const float* __restrict__ beta, int N) {
    int idx = blockIdx.x * blockDim.x + threadIdx.x;
    if (idx >= N * FDIM) return;
    int c = idx & (FDIM - 1);
    float invN = 1.0f / (float)N;
    float mu   = bsum[c] * invN;
    float var  = bsq[c] * invN - mu * mu;
    float y = gamma[c] * (out[idx] - mu) * rsqrtf(var + 1e-5f) + beta[c];
    out[idx] = (y > 0.f) ? y : 0.01f * y;
}

extern "C" void kernel_launch(void* const* d_in, const int* in_sizes, int n_in,
                              void* d_out, int out_size, void* d_ws, size_t ws_size,
                              hipStream_t stream) {
    const float*     x      = (const float*)d_in[0];
    const long long* eidx   = (const long long*)d_in[1];
    const float*     eattr  = (const float*)d_in[2];
    const float*     Wm     = (const float*)d_in[3];
    const float*     att_s  = (const float*)d_in[4];
    const float*     att_d  = (const float*)d_in[5];
    const float*     att_e  = (const float*)d_in[6];
    const float*     W_edge = (const float*)d_in[7];
    const float*     gamma  = (const float*)d_in[9];
    const float*     beta   = (const float*)d_in[10];

    const int N = in_sizes[0] / FDIM;
    const int E = in_sizes[1] / 2;
    const int K = E + N;                       // edges incl. self loops
    const long long* src = eidx;
    const long long* dst = eidx + E;

    // -------- workspace layout (floats) --------
    float* ws      = (float*)d_ws;
    float* h       = ws;                size_t o = (size_t)N * FDIM;
    float* a_src   = ws + o;            o += (size_t)N * HEADS;
    float* a_dst   = ws + o;            o += (size_t)N * HEADS;
    float* cnt     = ws + o;            o += (size_t)N;
    float* loopa   = ws + o;            o += (size_t)N * 2;   // lsum -> loop_attr
    float* pcoef   = ws + o;            o += 2 * HEADS;
    unsigned int* mEnc = (unsigned int*)(ws + o); o += (size_t)N * HEADS;
    float* ssum    = ws + o;            o += (size_t)N * HEADS;
    float* bsum    = ws + o;            o += FDIM;
    float* bsq     = ws + o;            o += FDIM;
    float* alphaE  = ws + o;            o += (size_t)K * HEADS;

    float* out = (float*)d_out;

    // -------- zero accumulators (graph-capture-safe async memsets) --------
    hipMemsetAsync(cnt,   0, (size_t)N * sizeof(float), stream);
    hipMemsetAsync(loopa, 0, (size_t)N * 2 * sizeof(float), stream);
    hipMemsetAsync(mEnc,  0, (size_t)N * HEADS * sizeof(unsigned int), stream);
    hipMemsetAsync(ssum,  0, (size_t)N * HEADS * sizeof(float), stream);
    hipMemsetAsync(bsum,  0, FDIM * sizeof(float), stream);
    hipMemsetAsync(bsq,   0, FDIM * sizeof(float), stream);
    hipMemsetAsync(out,   0, (size_t)N * FDIM * sizeof(float), stream);

    // -------- pipeline --------
    k_edge_cnt <<<(E + 255) / 256, 256, 0, stream>>>(dst, eattr, cnt, loopa, E);
    k_loop_attr<<<(N + 255) / 256, 256, 0, stream>>>(loopa, cnt, N);
    k_pcoef    <<<1, 64, 0, stream>>>(W_edge, att_e, pcoef);
    k_gemm_wmma<<<(N + 15) / 16, 256, 0, stream>>>(x, Wm, h, N);
    k_attdots  <<<(N * HEADS + 255) / 256, 256, 0, stream>>>(h, att_s, att_d,
                                                             a_src, a_dst, N);
    k_alpha_max<<<(K + 255) / 256, 256, 0, stream>>>(src, dst, eattr, loopa,
                                                     a_src, a_dst, pcoef,
                                                     alphaE, mEnc, E, K);
    k_exp_sum  <<<(K + 255) / 256, 256, 0, stream>>>(dst, alphaE, mEnc, ssum, E, K);
    k_scatter  <<<K, 128, 0, stream>>>(src, dst, alphaE, ssum, h, out, E);
    k_bn_stats <<<256, 128, 0, stream>>>(out, bsum, bsq, N);
    k_bn_apply <<<((size_t)N * FDIM + 255) / 256, 256, 0, stream>>>(out, bsum,
                                                                    bsq, gamma,
                                                                    beta, N);
}